// GraphNet3D_28819230556461
// MI455X (gfx1250) — compile-verified
//
#include <hip/hip_runtime.h>
#include <math.h>

typedef float v2f __attribute__((ext_vector_type(2)));
typedef float v8f __attribute__((ext_vector_type(8)));

#define K_CL   64
#define C_DIM  256
#define N_PTS  13824
#define TWO_C  512

// d_ws float layout:
//   Wt     [512][64]      @ 0       (B-operand: [inv_s2 ; -2*anchor*inv_s2], row=c', col=k)
//   invsig [64][256]      @ 32768
//   t3     [64]           @ 49152
//   wsum   [4][64]        @ 49216
//   wf     [4][64][256]   @ 49472
// total 115008 floats (~450 KB)

__global__ void zero_ws_kernel(float* __restrict__ p, int n) {
    int i = blockIdx.x * blockDim.x + threadIdx.x;
    if (i < n) p[i] = 0.f;
}

__global__ void prep_kernel(const float* __restrict__ anchor,
                            const float* __restrict__ sigma_param,
                            float* __restrict__ Wt,
                            float* __restrict__ invsig,
                            float* __restrict__ t3) {
    int idx = blockIdx.x * blockDim.x + threadIdx.x;   // 16384 = K*C
    int k = idx >> 8, c = idx & 255;
    float sp  = sigma_param[idx];
    float sig = 1.f / (1.f + __expf(-sp));
    float is2 = 1.f / (sig * sig);
    float a   = anchor[idx];
    Wt[c * K_CL + k]           = is2;              // pairs with f^2
    Wt[(C_DIM + c) * K_CL + k] = -2.f * a * is2;   // pairs with f
    invsig[idx] = 1.f / sig;
    atomicAdd(&t3[k], a * a * is2);
}

// One 64-row chunk of the 2C reduction; SQ selects the f^2 half at compile time
// so the hot loop has no per-element select.
template <bool SQ>
__device__ __forceinline__ void score_chunk(const float* __restrict__ xb,
                                            const float* __restrict__ Wt_chunk,
                                            float* __restrict__ ldsW,
                                            v8f (&acc)[4],
                                            int cmem, int n, int ln, int hi, int tid) {
    __syncthreads();
    const float4* src = (const float4*)Wt_chunk;
    float4* dst = (float4*)ldsW;
    #pragma unroll
    for (int q = 0; q < 4; ++q) dst[tid + 256 * q] = src[tid + 256 * q];
    __syncthreads();
    #pragma unroll 4
    for (int si = 0; si < 16; ++si) {
        const int crow = 4 * si + 2 * hi;          // reduction rows this lane feeds
        float f0 = xb[(size_t)(cmem + crow)     * N_PTS + n];
        float f1 = xb[(size_t)(cmem + crow + 1) * N_PTS + n];
        if (SQ) { f0 *= f0; f1 *= f1; }
        v2f a; a[0] = f0; a[1] = f1;
        #pragma unroll
        for (int j = 0; j < 4; ++j) {
            v2f bb;
            bb[0] = ldsW[crow       * K_CL + j * 16 + ln];
            bb[1] = ldsW[(crow + 1) * K_CL + j * 16 + ln];
            acc[j] = __builtin_amdgcn_wmma_f32_16x16x4_f32(
                false, a, false, bb, (short)0, acc[j], false, false);
        }
    }
}

// scores = -0.5*([f^2,f]·Wt^T + t3), softmax over K, emit soft_assign + w_sum partials.
// Block: 256 thr = 8 waves, each wave owns 16 points; grid (N/128, B).
__launch_bounds__(256, 2)
__global__ void assign_kernel(const float* __restrict__ x,
                              const float* __restrict__ Wt,
                              const float* __restrict__ t3,
                              float* __restrict__ wsum_g,
                              float* __restrict__ out2) {
    __shared__ float ldsW[64 * K_CL];   // 64 rows of Wt, 16 KB
    __shared__ float wsum_lds[K_CL];

    const int tid  = threadIdx.x;
    const int wave = tid >> 5, lane = tid & 31;
    const int ln = lane & 15, hi = lane >> 4;
    const int b  = blockIdx.y;
    const int nb = blockIdx.x * 128 + wave * 16;   // this wave's 16 points
    const int n  = nb + ln;                        // A-matrix row M = ln
    const float* xb = x + (size_t)b * C_DIM * N_PTS;

    if (tid < K_CL) wsum_lds[tid] = 0.f;

    v8f acc[4] = {};   // 4 k-tiles of 16x16 f32
    for (int chunk = 0; chunk < 4; ++chunk)        // rows 0..255: f^2 * inv_s2
        score_chunk<true >(xb, Wt + chunk * 64 * K_CL, ldsW, acc,
                           chunk * 64, n, ln, hi, tid);
    for (int chunk = 4; chunk < 8; ++chunk)        // rows 256..511: f * (-2*a*inv_s2)
        score_chunk<false>(xb, Wt + chunk * 64 * K_CL, ldsW, acc,
                           chunk * 64 - C_DIM, n, ln, hi, tid);

    // softmax over K=64 per point: rows spread over 16 lanes x 4 accumulators
    float t3v[4];
    #pragma unroll
    for (int j = 0; j < 4; ++j) t3v[j] = t3[j * 16 + ln];

    float w[4][8];
    #pragma unroll
    for (int v = 0; v < 8; ++v) {
        float m = -0.5f * (acc[0][v] + t3v[0]);
        #pragma unroll
        for (int j = 1; j < 4; ++j) m = fmaxf(m, -0.5f * (acc[j][v] + t3v[j]));
        m = fmaxf(m, __shfl_xor(m, 1)); m = fmaxf(m, __shfl_xor(m, 2));
        m = fmaxf(m, __shfl_xor(m, 4)); m = fmaxf(m, __shfl_xor(m, 8));
        float s = 0.f;
        #pragma unroll
        for (int j = 0; j < 4; ++j) {
            float e = __expf(-0.5f * (acc[j][v] + t3v[j]) - m);
            w[j][v] = e; s += e;
        }
        s += __shfl_xor(s, 1); s += __shfl_xor(s, 2);
        s += __shfl_xor(s, 4); s += __shfl_xor(s, 8);
        float inv = 1.f / s;
        #pragma unroll
        for (int j = 0; j < 4; ++j) w[j][v] *= inv;
    }

    // write soft_assign [B,K,N] (each lane owns 8 contiguous n's per k) + w_sum partials
    #pragma unroll
    for (int j = 0; j < 4; ++j) {
        const int k = j * 16 + ln;
        float* dstp = out2 + ((size_t)(b * K_CL + k)) * N_PTS + nb + hi * 8;
        ((float4*)dstp)[0] = make_float4(w[j][0], w[j][1], w[j][2], w[j][3]);
        ((float4*)dstp)[1] = make_float4(w[j][4], w[j][5], w[j][6], w[j][7]);
        float p = 0.f;
        #pragma unroll
        for (int v = 0; v < 8; ++v) p += w[j][v];
        p += __shfl_xor(p, 16);
        if (lane < 16) atomicAdd(&wsum_lds[k], p);
    }
    __syncthreads();
    if (tid < K_CL) atomicAdd(&wsum_g[b * K_CL + tid], wsum_lds[tid]);
}

// wf[b,k,c] = sum_n w[b,k,n] * x[b,c,n]; grid (N/512, B), block 256 = 8 waves.
// Each wave: 4 k-tiles x 2 c-tiles, reduction 512 n in steps of 4 (f32 WMMA K=4).
__launch_bounds__(256, 2)
__global__ void wf_kernel(const float* __restrict__ x,
                          const float* __restrict__ sa,
                          float* __restrict__ wf) {
    const int tid  = threadIdx.x;
    const int wave = tid >> 5, lane = tid & 31;
    const int ln = lane & 15, hi = lane >> 4;
    const int b  = blockIdx.y;
    const int nbase = blockIdx.x * 512;
    const int c0 = wave * 32;
    const float* xb  = x  + (size_t)b * C_DIM * N_PTS;
    const float* sab = sa + (size_t)b * K_CL  * N_PTS;

    v8f acc[4][2] = {};
    for (int s = 0; s < 128; ++s) {
        const int n0 = nbase + 4 * s + 2 * hi;     // two adjacent n's -> b64 load
        v2f amat[4], bmat[2];
        #pragma unroll
        for (int kt = 0; kt < 4; ++kt)
            amat[kt] = *(const v2f*)(sab + (size_t)(kt * 16 + ln) * N_PTS + n0);
        #pragma unroll
        for (int cj = 0; cj < 2; ++cj)
            bmat[cj] = *(const v2f*)(xb + (size_t)(c0 + cj * 16 + ln) * N_PTS + n0);
        #pragma unroll
        for (int kt = 0; kt < 4; ++kt)
            #pragma unroll
            for (int cj = 0; cj < 2; ++cj)
                acc[kt][cj] = __builtin_amdgcn_wmma_f32_16x16x4_f32(
                    false, amat[kt], false, bmat[cj], (short)0, acc[kt][cj], false, false);
    }
    #pragma unroll
    for (int kt = 0; kt < 4; ++kt)
        #pragma unroll
        for (int cj = 0; cj < 2; ++cj)
            #pragma unroll
            for (int v = 0; v < 8; ++v) {
                const int k = kt * 16 + (hi ? v + 8 : v);
                const int c = c0 + cj * 16 + ln;
                atomicAdd(&wf[((size_t)b * K_CL + k) * C_DIM + c], acc[kt][cj][v]);
            }
}

// nodes = (wf - anchor*wsum)/sigma/(wsum+eps); per-node L2 norm then per-batch L2 norm.
__global__ void finalize_kernel(const float* __restrict__ anchor,
                                const float* __restrict__ invsig,
                                const float* __restrict__ wf,
                                const float* __restrict__ wsum,
                                float* __restrict__ out1) {
    __shared__ float red;
    const int tid  = threadIdx.x;
    const int wave = tid >> 5, lane = tid & 31;
    const int b = blockIdx.x;
    if (tid == 0) red = 0.f;
    __syncthreads();

    float reg[8][8];
    #pragma unroll
    for (int i = 0; i < 8; ++i) {
        const int k  = wave * 8 + i;
        const float ws = wsum[b * K_CL + k];
        const float dn = 1.f / (ws + 1e-9f);
        float ss = 0.f;
        #pragma unroll
        for (int q = 0; q < 8; ++q) {
            const int idx = k * C_DIM + lane + 32 * q;
            float v = (wf[(size_t)b * K_CL * C_DIM + idx] - anchor[idx] * ws)
                      * invsig[idx] * dn;
            reg[i][q] = v; ss += v * v;
        }
        ss += __shfl_xor(ss, 1);  ss += __shfl_xor(ss, 2);
        ss += __shfl_xor(ss, 4);  ss += __shfl_xor(ss, 8);
        ss += __shfl_xor(ss, 16);
        const float inv1 = 1.f / fmaxf(sqrtf(ss), 1e-12f);
        #pragma unroll
        for (int q = 0; q < 8; ++q) reg[i][q] *= inv1;
        if (lane == 0) atomicAdd(&red, ss * inv1 * inv1);
    }
    __syncthreads();
    const float inv2 = 1.f / fmaxf(sqrtf(red), 1e-12f);
    #pragma unroll
    for (int i = 0; i < 8; ++i) {
        const int k = wave * 8 + i;
        #pragma unroll
        for (int q = 0; q < 8; ++q)
            out1[(size_t)b * K_CL * C_DIM + k * C_DIM + lane + 32 * q] = reg[i][q] * inv2;
    }
}

extern "C" void kernel_launch(void* const* d_in, const int* in_sizes, int n_in,
                              void* d_out, int out_size, void* d_ws, size_t ws_size,
                              hipStream_t stream) {
    (void)in_sizes; (void)n_in; (void)out_size; (void)ws_size;
    const float* x           = (const float*)d_in[0];
    const float* anchor      = (const float*)d_in[1];
    const float* sigma_param = (const float*)d_in[2];

    float* out  = (float*)d_out;
    float* out1 = out;                       // [B, C*K] normalized nodes (flat == [B,K,C] memory)
    float* out2 = out + 4 * K_CL * C_DIM;    // soft_assign [B, K, N]

    float* ws     = (float*)d_ws;
    float* Wt     = ws;             // 32768
    float* invsig = ws + 32768;     // 16384
    float* t3     = ws + 49152;     // 64
    float* wsum   = ws + 49216;     // 256
    float* wf     = ws + 49472;     // 65536

    // zero t3 + wsum + wf (contiguous: 65856 floats)
    zero_ws_kernel<<<(65856 + 255) / 256, 256, 0, stream>>>(t3, 65856);
    prep_kernel<<<K_CL, 256, 0, stream>>>(anchor, sigma_param, Wt, invsig, t3);
    assign_kernel<<<dim3(N_PTS / 128, 4), 256, 0, stream>>>(x, Wt, t3, wsum, out2);
    wf_kernel<<<dim3(N_PTS / 512, 4), 256, 0, stream>>>(x, out2, wf);
    finalize_kernel<<<4, 256, 0, stream>>>(anchor, invsig, wf, wsum, out1);
}